// HDSLinear_7687991459886
// MI455X (gfx1250) — compile-verified
//
#include <hip/hip_runtime.h>
#include <hip/hip_bf16.h>

// ---------------------------------------------------------------------------
// HDSLinear forward on MI455X (gfx1250), SWMMAC + async-LDS edition.
//   mask = exact 2:4 top-k(scores + gumbel(noise)) per group of 4 (K dim)
//   out[16384,4096] = X @ (W*mask)^T + bias
// GEMM: V_SWMMAC_F32_16X16X64_BF16 (weight = compressed 2:4 A operand).
// X tiles staged to LDS once per workgroup via GLOBAL_LOAD_ASYNC_TO_LDS_B128
// (ASYNCcnt), double-buffered; fragments read with ds_load_b128 (row stride
// padded to 72 bf16 -> 16 rows hit 16 distinct 4-bank groups).
// Roofline: memory-bound (~0.6 GB HBM -> ~25 us @ 23.3 TB/s); bf16 X
// (128 MB) is L2-resident, compact weight 16 MB + 2 MB indices.
// ---------------------------------------------------------------------------

typedef __bf16 bf16_t;
typedef __attribute__((ext_vector_type(8)))  __bf16 v8bf;
typedef __attribute__((ext_vector_type(16))) __bf16 v16bf;
typedef __attribute__((ext_vector_type(32))) __bf16 v32bf;
typedef __attribute__((ext_vector_type(8)))  float  v8f;
typedef __attribute__((ext_vector_type(4)))  float  f4;
typedef __attribute__((ext_vector_type(4)))  int    i4;

static constexpr int  BB   = 8;
static constexpr int  SS   = 2048;
static constexpr int  DIN  = 4096;
static constexpr int  DOUT = 4096;
static constexpr int  Mtot = BB * SS;       // 16384
static constexpr int  Kdim = DIN;           // 4096
static constexpr int  Ndim = DOUT;          // 4096
static constexpr int  KC   = Kdim / 2;      // 2048 compacted K per row
static constexpr int  NBLK = Kdim / 64;     // 64 K-blocks
static constexpr int  LSTR = 72;            // padded LDS row stride (bf16)

#if defined(__HIP_DEVICE_COMPILE__) && \
    __has_builtin(__builtin_amdgcn_global_load_async_to_lds_b128)
#define USE_ASYNC_LDS 1
#else
#define USE_ASYNC_LDS 0
#endif

__device__ __forceinline__ unsigned int f32_to_bf16_bits(float f) {
  unsigned int u = __float_as_uint(f);
  u += 0x7FFFu + ((u >> 16) & 1u);          // round-to-nearest-even
  return (u >> 16) & 0xFFFFu;
}

template <int N>
__device__ __forceinline__ void wait_async() {
#if USE_ASYNC_LDS
#if __has_builtin(__builtin_amdgcn_s_wait_asynccnt)
  __builtin_amdgcn_s_wait_asynccnt(N);      // immediate via template param
#else
  if (N == 0)
    asm volatile("s_wait_asynccnt 0x0" ::: "memory");
  else
    asm volatile("s_wait_asynccnt 0x4" ::: "memory");
#endif
#endif
}

// ---------------------------------------------------------------------------
// Kernel 1: 2:4 selection + compaction. One thread per (out row, 64-K block)
// = 16 groups of 4. Emits:
//   Wc  [DOUT, KC]      compacted bf16 kept weights (2 per group, k-order)
//   Widx[DOUT, NBLK, 2] index words, hardware layout (ISA 7.12.4):
//     word h holds groups h*8..h*8+7; field f=g%8:
//     bits[4f+1:4f]=idx0, bits[4f+3:4f+2]=idx1  (idx0 < idx1)
// ---------------------------------------------------------------------------
__global__ __launch_bounds__(256) void hds_sparse_prep_kernel(
    const float* __restrict__ weight, const float* __restrict__ scores,
    const float* __restrict__ noise_u, unsigned short* __restrict__ Wc,
    unsigned int* __restrict__ Widx) {
  const int t   = blockIdx.x * blockDim.x + threadIdx.x;   // row*NBLK + blk
  const int o   = t / NBLK;
  const int blk = t % NBLK;
  if (o >= DOUT) return;

  const long long gbase = (long long)o * Kdim + blk * 64;
  unsigned short* wcrow = Wc + (size_t)o * KC + blk * 32;
  unsigned int idxw[2] = {0u, 0u};

#pragma unroll
  for (int g = 0; g < 16; ++g) {
    const long long off = gbase + g * 4;
    const f4 s = *(const f4*)(scores + off);
    const f4 u = *(const f4*)(noise_u + off);
    const f4 w = *(const f4*)(weight + off);

    float y[4];
#pragma unroll
    for (int j = 0; j < 4; ++j) {
      const float gb = -__logf(-__logf(u[j] + 1e-10f) + 1e-10f);
      y[j] = s[j] + gb;                      // softmax monotone: rank on y
    }
    int i0 = 0;                              // ties -> lowest index
#pragma unroll
    for (int j = 1; j < 4; ++j) if (y[j] > y[i0]) i0 = j;
    int i1 = -1;
#pragma unroll
    for (int j = 0; j < 4; ++j) {
      if (j == i0) continue;
      if (i1 < 0 || y[j] > y[i1]) i1 = j;
    }
    const int lo = i0 < i1 ? i0 : i1;        // hardware wants idx0 < idx1
    const int hi = i0 < i1 ? i1 : i0;

    const unsigned int pair =
        f32_to_bf16_bits(w[lo]) | (f32_to_bf16_bits(w[hi]) << 16);
    *(unsigned int*)(wcrow + 2 * g) = pair;

    const int h = g >> 3, f = g & 7;
    idxw[h] |= ((unsigned)lo | ((unsigned)hi << 2)) << (4 * f);
  }
  unsigned long long packed =
      (unsigned long long)idxw[0] | ((unsigned long long)idxw[1] << 32);
  *(unsigned long long*)(Widx + (size_t)t * 2) = packed;
}

// ---------------------------------------------------------------------------
// Kernel 2: x fp32 -> bf16 (8 elements / thread)
// ---------------------------------------------------------------------------
__global__ __launch_bounds__(256) void hds_x_to_bf16_kernel(
    const float* __restrict__ x, unsigned short* __restrict__ Xb) {
  const long long t =
      ((long long)blockIdx.x * blockDim.x + threadIdx.x) * 8;
  const f4 a = *(const f4*)(x + t);
  const f4 b = *(const f4*)(x + t + 4);
  unsigned long long p0 = 0ull, p1 = 0ull;
#pragma unroll
  for (int j = 0; j < 4; ++j) {
    p0 |= (unsigned long long)f32_to_bf16_bits(a[j]) << (16 * j);
    p1 |= (unsigned long long)f32_to_bf16_bits(b[j]) << (16 * j);
  }
  unsigned long long* dst = (unsigned long long*)(Xb + t);
  dst[0] = p0;
  dst[1] = p1;
}

// ---------------------------------------------------------------------------
// Kernel 3: SWMMAC GEMM with double-buffered async X staging.
// Workgroup: 256 thr = 8 wave32 -> 128(out) x 128(m) tile; wave: 32x64.
// Per K-block (64 dense K):
//   X tile 128x64 bf16 staged to LDS (async b128, 4 chunks/thread),
//   A/idx (compressed weight) direct global, 8 swmmac per wave.
// ---------------------------------------------------------------------------
__global__ __launch_bounds__(256) void hds_swmmac_gemm_kernel(
    const bf16_t* __restrict__ Xb, const bf16_t* __restrict__ Wc,
    const unsigned int* __restrict__ Widx, const float* __restrict__ bias,
    float* __restrict__ out) {
  __shared__ __align__(16) unsigned short lds_x[2][128 * LSTR];

  const int tid  = threadIdx.x;
  const int lane = tid & 31;
  const int wid  = tid >> 5;
  const int wr   = wid & 3;           // wave row (out dir): 4 x 32
  const int wc   = wid >> 2;          // wave col (m   dir): 2 x 64
  const int l15  = lane & 15;
  const int lhi  = lane >> 4;

  const int tileO = blockIdx.x * 128;     // out (N) tile
  const int tileM = blockIdx.y * 128;     // m   (M) tile

  // --- A (compressed weight) + index pointers ---------------------------
  const bf16_t* aptr[2];
  const unsigned int* iptr[2];
#pragma unroll
  for (int i = 0; i < 2; ++i) {
    const int row = tileO + wr * 32 + i * 16 + l15;
    aptr[i] = Wc + (size_t)row * KC + lhi * 8;
    iptr[i] = Widx + (size_t)row * (NBLK * 2) + lhi;
  }
  // --- B fragment LDS offsets (elements) --------------------------------
  int boff[4];
#pragma unroll
  for (int j = 0; j < 4; ++j)
    boff[j] = (wc * 64 + j * 16 + l15) * LSTR + lhi * 16;

  v8f acc[2][4];
#pragma unroll
  for (int i = 0; i < 2; ++i)
#pragma unroll
    for (int j = 0; j < 4; ++j) acc[i][j] = (v8f)(0.0f);

  // stage helper: copy X tile [128 x 64] of block `blk` into lds buffer
  auto stage = [&](int buf, int blk) {
    unsigned short* lb = &lds_x[buf][0];
#pragma unroll
    for (int q = 0; q < 4; ++q) {
      const int chunk = tid * 4 + q;          // 0..1023 16B chunks
      const int row   = chunk >> 3;           // 8 chunks per 64-elem row
      const int cc    = (chunk & 7) * 8;
      const bf16_t* gp =
          Xb + (size_t)(tileM + row) * Kdim + blk * 64 + cc;
      unsigned short* lp = lb + row * LSTR + cc;
#if USE_ASYNC_LDS
      __builtin_amdgcn_global_load_async_to_lds_b128(
          (__attribute__((address_space(1))) i4*)(i4*)gp,
          (__attribute__((address_space(3))) i4*)(i4*)lp, 0, 0);
#else
      *(v8bf*)lp = *(const v8bf*)gp;
#endif
    }
  };

  stage(0, 0);

  for (int blk = 0; blk < NBLK; ++blk) {
    const int kc = blk * 32;                 // compacted-K offset
    const unsigned short* cur = &lds_x[blk & 1][0];

    if (blk + 1 < NBLK) {
      stage((blk + 1) & 1, blk + 1);         // overlap next tile's DMA
      wait_async<4>();                       // oldest 4 (cur tile) done
      __builtin_prefetch(aptr[0] + kc + 32, 0, 3);  // next weight block
      __builtin_prefetch(aptr[1] + kc + 32, 0, 3);
    } else {
      wait_async<0>();
    }
    __syncthreads();                         // cur tile visible to all waves

    v16bf a[2];
    int   idx[2];
#pragma unroll
    for (int i = 0; i < 2; ++i) {
      const v8bf lo = *(const v8bf*)(aptr[i] + kc);
      const v8bf hi = *(const v8bf*)(aptr[i] + kc + 16);
      a[i] = __builtin_shufflevector(lo, hi, 0, 1, 2, 3, 4, 5, 6, 7,
                                     8, 9, 10, 11, 12, 13, 14, 15);
      idx[i] = (int)iptr[i][blk * 2];
    }

    v32bf b[4];
#pragma unroll
    for (int j = 0; j < 4; ++j) {
      const v16bf blo = *(const v16bf*)(cur + boff[j]);        // K 0..15(+16)
      const v16bf bhi = *(const v16bf*)(cur + boff[j] + 32);   // K 32..47(+16)
      b[j] = __builtin_shufflevector(
          blo, bhi, 0, 1, 2, 3, 4, 5, 6, 7, 8, 9, 10, 11, 12, 13, 14, 15,
          16, 17, 18, 19, 20, 21, 22, 23, 24, 25, 26, 27, 28, 29, 30, 31);
    }

#pragma unroll
    for (int i = 0; i < 2; ++i)
#pragma unroll
      for (int j = 0; j < 4; ++j)
        acc[i][j] = __builtin_amdgcn_swmmac_f32_16x16x64_bf16(
            false, a[i], false, b[j], acc[i][j], idx[i], false, false);

    __syncthreads();     // protect buffer reuse by next iteration's stage
  }

  // Epilogue: D[o,m]; lane -> m = (l&15), o = (l>>4)*8 + r (consecutive)
#pragma unroll
  for (int i = 0; i < 2; ++i) {
    const int o0 = tileO + wr * 32 + i * 16 + lhi * 8;
    const f4 bz0 = *(const f4*)(bias + o0);
    const f4 bz1 = *(const f4*)(bias + o0 + 4);
#pragma unroll
    for (int j = 0; j < 4; ++j) {
      const int m = tileM + wc * 64 + j * 16 + l15;
      float* op = out + (size_t)m * Ndim + o0;
      f4 r0, r1;
#pragma unroll
      for (int r = 0; r < 4; ++r) {
        r0[r] = acc[i][j][r] + bz0[r];
        r1[r] = acc[i][j][r + 4] + bz1[r];
      }
      *(f4*)op       = r0;
      *(f4*)(op + 4) = r1;
    }
  }
}

// ---------------------------------------------------------------------------
// Launch. d_in = {x, weight, bias, scores, noise_u} (fp32).
// Workspace: [Xb bf16 128MB][Wc bf16 16MB][Widx u32 2MB] = 146 MB.
// ---------------------------------------------------------------------------
extern "C" void kernel_launch(void* const* d_in, const int* in_sizes, int n_in,
                              void* d_out, int out_size, void* d_ws,
                              size_t ws_size, hipStream_t stream) {
  const float* x       = (const float*)d_in[0];
  const float* weight  = (const float*)d_in[1];
  const float* bias    = (const float*)d_in[2];
  const float* scores  = (const float*)d_in[3];
  const float* noise_u = (const float*)d_in[4];
  float* out = (float*)d_out;

  unsigned short* Xb = (unsigned short*)d_ws;
  unsigned short* Wc = Xb + (size_t)Mtot * Kdim;
  unsigned int*   Wi = (unsigned int*)(Wc + (size_t)DOUT * KC);

  {
    const int total  = DOUT * NBLK;            // 262144
    const int blocks = (total + 255) / 256;
    hds_sparse_prep_kernel<<<blocks, 256, 0, stream>>>(weight, scores, noise_u,
                                                       Wc, Wi);
  }
  {
    const long long elems = (long long)Mtot * Kdim;
    const int blocks = (int)(elems / (256 * 8));
    hds_x_to_bf16_kernel<<<blocks, 256, 0, stream>>>(x, Xb);
  }
  {
    dim3 grid(DOUT / 128, Mtot / 128);         // (32, 128)
    hds_swmmac_gemm_kernel<<<grid, 256, 0, stream>>>(
        (const bf16_t*)Xb, (const bf16_t*)Wc, Wi, bias, out);
  }
}